// Seq_56899726738098
// MI455X (gfx1250) — compile-verified
//
#include <hip/hip_runtime.h>
#include <hip/hip_bf16.h>
#include <math.h>

// Problem constants (from reference setup_inputs)
#define B_  4
#define S_  512
#define SC_ 128
#define MC_ 64
#define R_  32
#define O_  64
#define H_  (4 * O_)          // 256
#define INV_DENOM 0.17677669529663687f  // 1/sqrt(32)
#define LN_EPS 1e-5f

typedef float v2f __attribute__((ext_vector_type(2)));
typedef float v8f __attribute__((ext_vector_type(8)));

// ---------------------------------------------------------------------------
// Kernel 1: q = seq@Wq+bq, k = seq@Wk+bk.
// One 32-lane group per (b,s, q-or-k) row; lane r computes output element r.
// Weight reads W[c*R + r] are coalesced across lanes; seq value is broadcast.
// ---------------------------------------------------------------------------
__global__ void qk_kernel(const float* __restrict__ seq,
                          const float* __restrict__ Wq, const float* __restrict__ bq,
                          const float* __restrict__ Wk, const float* __restrict__ bk,
                          float* __restrict__ q, float* __restrict__ k) {
    int group = blockIdx.x * (blockDim.x / 32) + (threadIdx.x >> 5); // 0..B*S*2-1
    int lane  = threadIdx.x & 31;
    int isK   = group & 1;
    int bs    = group >> 1;                       // 0..B*S-1
    const float* W    = isK ? Wk : Wq;
    const float* bias = isK ? bk : bq;
    const float* srow = seq + (size_t)bs * SC_;
    float acc = bias[lane];
    #pragma unroll 4
    for (int c = 0; c < SC_; ++c)
        acc = fmaf(srow[c], W[c * R_ + lane], acc);
    (isK ? k : q)[(size_t)bs * R_ + lane] = acc;
}

// ---------------------------------------------------------------------------
// Kernel 2: logits = q·k^T / sqrt(R), softmax over t. One block per (b,s).
// 256 threads, each owns 2 t-values; block-wide max/sum reductions in LDS.
// ---------------------------------------------------------------------------
__global__ void attn_kernel(const float* __restrict__ q,
                            const float* __restrict__ k,
                            float* __restrict__ attn) {
    __shared__ float qs[R_];
    __shared__ float red[256];
    int bs = blockIdx.x;            // 0..B*S-1
    int b  = bs / S_;
    int tid = threadIdx.x;

    if (tid < R_) qs[tid] = q[(size_t)bs * R_ + tid];
    __syncthreads();

    float lg[2];
    #pragma unroll
    for (int i = 0; i < 2; ++i) {
        int t = tid + i * 256;
        const float* krow = k + ((size_t)b * S_ + t) * R_;
        float d = 0.f;
        #pragma unroll
        for (int r = 0; r < R_; ++r) d = fmaf(qs[r], krow[r], d);
        lg[i] = d * INV_DENOM;
    }

    // block max
    float mloc = fmaxf(lg[0], lg[1]);
    red[tid] = mloc;  __syncthreads();
    for (int off = 128; off > 0; off >>= 1) {
        if (tid < off) red[tid] = fmaxf(red[tid], red[tid + off]);
        __syncthreads();
    }
    float mx = red[0];  __syncthreads();

    float e0 = __expf(lg[0] - mx);
    float e1 = __expf(lg[1] - mx);
    red[tid] = e0 + e1;  __syncthreads();
    for (int off = 128; off > 0; off >>= 1) {
        if (tid < off) red[tid] += red[tid + off];
        __syncthreads();
    }
    float inv = 1.f / red[0];

    float* arow = attn + (size_t)bs * S_;
    arow[tid]       = e0 * inv;
    arow[tid + 256] = e1 * inv;
}

// ---------------------------------------------------------------------------
// Kernel 3 (bandwidth-dominant): m[b,s,c] = sum_t attn[b,s,t] * map_[b,c,s,t].
// One block per (b,s); attn row cached in LDS and reused across all 64 c.
// Each wave handles 8 c-rows; lanes stream 512 contiguous floats per row with
// float4 (b128) loads, then wave-reduce. map_ is read exactly once (268 MB).
// ---------------------------------------------------------------------------
__global__ void mix_kernel(const float* __restrict__ map_,
                           const float* __restrict__ attn,
                           float* __restrict__ m) {
    __shared__ __align__(16) float a[S_];
    int bs = blockIdx.x;
    int b  = bs / S_;
    int s  = bs - b * S_;
    int tid = threadIdx.x;

    // 256 threads load 512 floats (float2 each), coalesced
    {
        float2 v = ((const float2*)(attn + (size_t)bs * S_))[tid];
        *(float2*)(a + tid * 2) = v;
    }
    __syncthreads();

    int wave = tid >> 5;
    int lane = tid & 31;

    #pragma unroll
    for (int i = 0; i < 8; ++i) {
        int c = wave * 8 + i;
        const float* row = map_ + (((size_t)b * MC_ + c) * S_ + s) * (size_t)S_;
        float acc = 0.f;
        #pragma unroll
        for (int j = 0; j < 4; ++j) {
            int t0 = j * 128 + lane * 4;
            float4 mv = *(const float4*)(row + t0);
            float4 av = *(const float4*)(a + t0);
            acc = fmaf(mv.x, av.x, acc);
            acc = fmaf(mv.y, av.y, acc);
            acc = fmaf(mv.z, av.z, acc);
            acc = fmaf(mv.w, av.w, acc);
        }
        // wave32 reduction
        #pragma unroll
        for (int off = 16; off > 0; off >>= 1)
            acc += __shfl_xor(acc, off, 32);
        if (lane == 0) m[(size_t)bs * MC_ + c] = acc;
    }
}

// ---------------------------------------------------------------------------
// Kernel 4: pooled = m@Wv+bv -> layernorm -> relu(score@W1+b1) -> @W2+b2.
// One wave per 16-row s-tile; all three GEMMs via V_WMMA_F32_16X16X4_F32.
//
// f32 WMMA lane layouts (wave32):
//  A (16x4):  lanes 0-15 -> M=lane, {VGPR0,VGPR1}={K0,K1}; lanes 16-31 {K2,K3}
//  B (4x16):  lanes 0-15 -> N=lane, {VGPR0,VGPR1}={K0,K1}; lanes 16-31 {K2,K3}
//  C/D 16x16: VGPR v, lanes 0-15 -> (M=v,   N=lane),
//                      lanes16-31 -> (M=v+8, N=lane-16)
// ---------------------------------------------------------------------------
#define AS_STRIDE 66   // even -> float2-aligned LDS rows, avoids bank clash
#define HS_STRIDE 260

__global__ void head_kernel(const float* __restrict__ m,
                            const float* __restrict__ Wv, const float* __restrict__ bv,
                            const float* __restrict__ gamma, const float* __restrict__ beta,
                            const float* __restrict__ W1, const float* __restrict__ b1,
                            const float* __restrict__ W2, const float* __restrict__ b2,
                            float* __restrict__ out) {
    __shared__ __align__(16) float As[16 * AS_STRIDE];   // m tile
    __shared__ __align__(16) float Ps[16 * AS_STRIDE];   // pooled / score tile
    __shared__ __align__(16) float Hs[16 * HS_STRIDE];   // MLP hidden tile

    int lane = threadIdx.x;            // 0..31 (one wave)
    int row0 = blockIdx.x * 16;        // flattened (b*S+s) row base

    // Stage m tile [16][64] into LDS (padded stride)
    const float* msrc = m + (size_t)row0 * MC_;
    for (int r = 0; r < 16; ++r) {
        As[r * AS_STRIDE + lane]      = msrc[r * MC_ + lane];
        As[r * AS_STRIDE + 32 + lane] = msrc[r * MC_ + 32 + lane];
    }
    __syncthreads();

    const int arow = lane & 15;
    const int asel = (lane >> 4) << 1;       // 0 or 2
    const int ncol = lane & 15;
    const int mbase = (lane >> 4) << 3;      // D rows: mbase + v

    // ---- GEMM 1: pooled = m @ Wv + bv  (K=64, N=64) ----
    #pragma unroll
    for (int nt = 0; nt < 4; ++nt) {
        v8f acc;
        float bb = bv[nt * 16 + ncol];
        #pragma unroll
        for (int i = 0; i < 8; ++i) acc[i] = bb;
        for (int k0 = 0; k0 < MC_; k0 += 4) {
            v2f av = *(const v2f*)(&As[arow * AS_STRIDE + k0 + asel]);
            v2f bvv;
            bvv.x = Wv[(k0 + asel)     * MC_ + nt * 16 + ncol];
            bvv.y = Wv[(k0 + asel + 1) * MC_ + nt * 16 + ncol];
            acc = __builtin_amdgcn_wmma_f32_16x16x4_f32(false, av, false, bvv,
                                                        (short)0, acc, false, false);
        }
        #pragma unroll
        for (int i = 0; i < 8; ++i)
            Ps[(mbase + i) * AS_STRIDE + nt * 16 + ncol] = acc[i];
    }
    __syncthreads();

    // ---- LayerNorm over c (lanes 0-15 own one row each) ----
    if (lane < 16) {
        float* prow = Ps + lane * AS_STRIDE;
        float mu = 0.f;
        for (int c = 0; c < MC_; ++c) mu += prow[c];
        mu *= (1.f / MC_);
        float var = 0.f;
        for (int c = 0; c < MC_; ++c) { float d = prow[c] - mu; var = fmaf(d, d, var); }
        var *= (1.f / MC_);
        float rs = rsqrtf(var + LN_EPS);
        for (int c = 0; c < MC_; ++c)
            prow[c] = (prow[c] - mu) * rs * gamma[c] + beta[c];
    }
    __syncthreads();

    // ---- GEMM 2: h = relu(score @ W1 + b1)  (K=64, N=256) ----
    for (int nt = 0; nt < 16; ++nt) {
        v8f acc;
        float bb = b1[nt * 16 + ncol];
        #pragma unroll
        for (int i = 0; i < 8; ++i) acc[i] = bb;
        for (int k0 = 0; k0 < MC_; k0 += 4) {
            v2f av = *(const v2f*)(&Ps[arow * AS_STRIDE + k0 + asel]);
            v2f bvv;
            bvv.x = W1[(k0 + asel)     * H_ + nt * 16 + ncol];
            bvv.y = W1[(k0 + asel + 1) * H_ + nt * 16 + ncol];
            acc = __builtin_amdgcn_wmma_f32_16x16x4_f32(false, av, false, bvv,
                                                        (short)0, acc, false, false);
        }
        #pragma unroll
        for (int i = 0; i < 8; ++i)
            Hs[(mbase + i) * HS_STRIDE + nt * 16 + ncol] = fmaxf(acc[i], 0.f);
    }
    __syncthreads();

    // ---- GEMM 3: out = h @ W2 + b2  (K=256, N=64) ----
    #pragma unroll
    for (int nt = 0; nt < 4; ++nt) {
        v8f acc;
        float bb = b2[nt * 16 + ncol];
        #pragma unroll
        for (int i = 0; i < 8; ++i) acc[i] = bb;
        for (int k0 = 0; k0 < H_; k0 += 4) {
            v2f av = *(const v2f*)(&Hs[arow * HS_STRIDE + k0 + asel]);
            v2f bvv;
            bvv.x = W2[(k0 + asel)     * O_ + nt * 16 + ncol];
            bvv.y = W2[(k0 + asel + 1) * O_ + nt * 16 + ncol];
            acc = __builtin_amdgcn_wmma_f32_16x16x4_f32(false, av, false, bvv,
                                                        (short)0, acc, false, false);
        }
        #pragma unroll
        for (int i = 0; i < 8; ++i)
            out[(size_t)(row0 + mbase + i) * O_ + nt * 16 + ncol] = acc[i];
    }
}

// ---------------------------------------------------------------------------
extern "C" void kernel_launch(void* const* d_in, const int* in_sizes, int n_in,
                              void* d_out, int out_size, void* d_ws, size_t ws_size,
                              hipStream_t stream) {
    (void)in_sizes; (void)n_in; (void)out_size; (void)ws_size;

    const float* map_  = (const float*)d_in[0];
    const float* seq   = (const float*)d_in[1];
    const float* Wq    = (const float*)d_in[2];
    const float* bq    = (const float*)d_in[3];
    const float* Wk    = (const float*)d_in[4];
    const float* bk    = (const float*)d_in[5];
    const float* Wv    = (const float*)d_in[6];
    const float* bv    = (const float*)d_in[7];
    const float* gamma = (const float*)d_in[8];
    const float* beta  = (const float*)d_in[9];
    const float* W1    = (const float*)d_in[10];
    const float* b1    = (const float*)d_in[11];
    const float* W2    = (const float*)d_in[12];
    const float* b2    = (const float*)d_in[13];
    float* out = (float*)d_out;

    // Workspace partition (floats): q, k, attn, m
    float* ws   = (float*)d_ws;
    float* q    = ws;                                   // B*S*R  = 65536
    float* k    = q    + (size_t)B_ * S_ * R_;          // 65536
    float* attn = k    + (size_t)B_ * S_ * R_;          // B*S*S = 1048576
    float* mbuf = attn + (size_t)B_ * S_ * S_;          // B*S*MC = 131072

    qk_kernel  <<<(B_ * S_ * 2) / 8, 256, 0, stream>>>(seq, Wq, bq, Wk, bk, q, k);
    attn_kernel<<<B_ * S_,          256, 0, stream>>>(q, k, attn);
    mix_kernel <<<B_ * S_,          256, 0, stream>>>(map_, attn, mbuf);
    head_kernel<<<(B_ * S_) / 16,    32, 0, stream>>>(mbuf, Wv, bv, gamma, beta,
                                                      W1, b1, W2, b2, out);
}